// RoiUpsample_27178553049409
// MI455X (gfx1250) — compile-verified
//
#include <hip/hip_runtime.h>
#include <hip/hip_bf16.h>

#ifndef __has_builtin
#define __has_builtin(x) 0
#endif

// Problem constants (fixed by the reference).
#define N_ 8
#define M_ 16384
#define C_ 256
#define H_ 200
#define W_ 200
#define HW_ (H_ * W_)

typedef __attribute__((ext_vector_type(2))) float v2f;
typedef __attribute__((ext_vector_type(8))) float v8f;
typedef __attribute__((ext_vector_type(4))) int   v4i;

// ---------------------------------------------------------------------------
// Kernel 1: per-roi bilinear scatter-add with atomics.
//   DIRECT=false : accumulate into NHWC scratch ws[n][hw][c]  (contiguous atomics)
//   DIRECT=true  : accumulate straight into NCHW d_out (fallback if ws too small)
// One wave (32 lanes) per roi; 8 rois per 256-thread block. The block's
// 8 KB of feature data is staged global->LDS with the CDNA5 async path.
// ---------------------------------------------------------------------------
template <bool DIRECT>
__global__ __launch_bounds__(256) void roi_scatter_kernel(
    const float* __restrict__ center,   // [N*M][2]
    const float* __restrict__ feat,     // [N*M][C]
    float* __restrict__ acc)            // NHWC scratch or NCHW out
{
    __shared__ float smem[8 * C_];      // 8 rois x 256 ch = 8 KB

    const int tid  = threadIdx.x;
    const int wave = tid >> 5;
    const int lane = tid & 31;
    const size_t roiBase = (size_t)blockIdx.x * 8;

    // ---- Stage 8 rois x 256 floats (contiguous 8 KB) into LDS ------------
#if defined(__HIP_DEVICE_COMPILE__) && __has_builtin(__builtin_amdgcn_global_load_async_to_lds_b128)
    {
        // 256 threads x 16 B = 4 KB per issue; two issues cover 8 KB.
        v4i* g0 = (v4i*)(feat + roiBase * C_ + (size_t)tid * 4);
        v4i* g1 = (v4i*)(feat + roiBase * C_ + 1024 + (size_t)tid * 4);
        v4i* l0 = (v4i*)&smem[tid * 4];
        v4i* l1 = (v4i*)&smem[1024 + tid * 4];
        __builtin_amdgcn_global_load_async_to_lds_b128(g0, l0, 0, 0);
        __builtin_amdgcn_global_load_async_to_lds_b128(g1, l1, 0, 0);
#if __has_builtin(__builtin_amdgcn_s_wait_asynccnt)
        __builtin_amdgcn_s_wait_asynccnt(0);
#else
        asm volatile("s_wait_asynccnt 0x0" ::: "memory");
#endif
    }
#else
    {
        const float4* g = (const float4*)(feat + roiBase * C_);
        float4 t0 = g[tid];
        float4 t1 = g[tid + 256];
        ((float4*)smem)[tid]       = t0;
        ((float4*)smem)[tid + 256] = t1;
    }
#endif
    __syncthreads();

    // ---- Per-roi scalars (computed redundantly by all 32 lanes) ----------
    const size_t roi = roiBase + wave;
    const int n = (int)(roi >> 14);                 // roi / M_
    const float cx = center[roi * 2 + 0];
    const float cy = center[roi * 2 + 1];
    const float flx = floorf(cx), fly = floorf(cy);
    const int fx = (int)flx, fy = (int)fly;
    const int cxi = fx + 1,  cyi = fy + 1;
    const float xp = cx - flx, yp = cy - fly;
    const float oxp = 1.0f - xp, oyp = 1.0f - yp;
    const float vx = xp * xp + oxp * oxp;
    const float vy = yp * yp + oyp * oyp;
    const float s = 0.25f / (vx * vy);

    const float wgt[4] = { oxp * oyp, oxp * yp, xp * oyp, xp * yp };
    const int   px[4]  = { fx, fx, cxi, cxi };
    const int   py[4]  = { fy, cyi, fy, cyi };

    bool  ok[4];
    size_t base[4];
#pragma unroll
    for (int j = 0; j < 4; ++j) {
        ok[j] = (px[j] < H_) && (py[j] < H_);       // reference masks both with H
        const int idx = px[j] * W_ + py[j];
        base[j] = DIRECT ? ((size_t)n * C_ * HW_ + (size_t)idx)
                         : (((size_t)n * HW_ + (size_t)idx) * C_);
    }

    const float* f = smem + wave * C_;

    // Lane handles channels c = lane + 32k -> atomics coalesced across the wave.
#pragma unroll
    for (int k = 0; k < C_ / 32; ++k) {
        const int c = lane + 32 * k;
        const float v = s * f[c];
#pragma unroll
        for (int j = 0; j < 4; ++j) {
            if (ok[j]) {
                const size_t off = DIRECT ? (base[j] + (size_t)c * HW_)
                                          : (base[j] + (size_t)c);
                atomicAdd(acc + off, wgt[j] * v);
            }
        }
    }
}

// ---------------------------------------------------------------------------
// Kernel 2: NHWC -> NCHW transpose of the accumulator.
// Per 16(hw) x 16(c) f32 tile: redistribute with V_WMMA_F32_16X16X4_F32 and
// identity B chunks (A-layout -> C/D-layout = cross-lane transpose), giving
// coalesced loads and stores on the matrix pipe. One wave per tile-pair.
// ---------------------------------------------------------------------------
__global__ __launch_bounds__(256) void transpose_nhwc_to_nchw_kernel(
    const float* __restrict__ acc,      // [N][HW][C]
    float* __restrict__ out)            // [N][C][HW]
{
    const int tid  = threadIdx.x;
    const int wave = tid >> 5;
    const int lane = tid & 31;
    const int half = lane >> 4;         // 0: lanes 0-15, 1: lanes 16-31
    const int l    = lane & 15;

    const int strip = blockIdx.x;       // one (n, hw0) strip per block
    const int n   = strip / (HW_ / 16);
    const int hw0 = (strip % (HW_ / 16)) * 16;

#pragma unroll
    for (int t = 0; t < 2; ++t) {
        const int c0 = (wave + t * 8) * 16;   // 8 waves x 2 -> all 16 c-tiles
        const float* src = acc + ((size_t)n * HW_ + hw0) * C_ + c0;

#if defined(__HIP_DEVICE_COMPILE__) && __has_builtin(__builtin_amdgcn_wmma_f32_16x16x4_f32)
        v8f d = {};
#pragma unroll
        for (int b = 0; b < 4; ++b) {
            // A_b[i][k] = src[(4b+k)*C + i]; lane holds i=l, k rows per ISA
            // 16x4 f32 A layout: lanes 0-15 -> K=0,1 ; lanes 16-31 -> K=2,3.
            const int k0 = 4 * b + (half ? 2 : 0);
            const int k1 = 4 * b + (half ? 3 : 1);
            v2f a;
            a.x = src[(size_t)k0 * C_ + l];
            a.y = src[(size_t)k1 * C_ + l];
            // B_b[k][j] = delta(4b+k, j): identity slice in 4x16 B layout.
            v2f bb;
            bb.x = (l == k0) ? 1.0f : 0.0f;
            bb.y = (l == k1) ? 1.0f : 0.0f;
            d = __builtin_amdgcn_wmma_f32_16x16x4_f32(
                false, a, false, bb, (short)0, d, false, false);
        }
        // D layout: VGPR v holds D[M = v + 8*half][N = l];  M=c_rel, N=hw_rel.
        float* dst = out + ((size_t)(n * C_ + c0 + 8 * half)) * HW_ + hw0 + l;
#pragma unroll
        for (int v = 0; v < 8; ++v)
            dst[(size_t)v * HW_] = d[v];
#else
        // Fallback: scalar transpose of the 16x16 tile.
        for (int r = lane; r < 256; r += 32) {
            const int ci = r >> 4, hwi = r & 15;
            out[((size_t)(n * C_ + c0 + ci)) * HW_ + hw0 + hwi] =
                src[(size_t)hwi * C_ + ci];
        }
#endif
    }
}

// ---------------------------------------------------------------------------
extern "C" void kernel_launch(void* const* d_in, const int* in_sizes, int n_in,
                              void* d_out, int out_size, void* d_ws, size_t ws_size,
                              hipStream_t stream) {
    const float* center = (const float*)d_in[0];   // (N, M, 2)  f32
    const float* feat   = (const float*)d_in[1];   // (N, M, C)  f32
    float* out = (float*)d_out;                    // (N, C, H, W) f32

    const size_t accBytes = (size_t)N_ * HW_ * C_ * sizeof(float);
    const int nBlocksScatter = (N_ * M_) / 8;      // 8 rois per block

    if (ws_size >= accBytes) {
        float* acc = (float*)d_ws;
        (void)hipMemsetAsync(acc, 0, accBytes, stream);
        roi_scatter_kernel<false><<<nBlocksScatter, 256, 0, stream>>>(center, feat, acc);
        transpose_nhwc_to_nchw_kernel<<<N_ * (HW_ / 16), 256, 0, stream>>>(acc, out);
    } else {
        (void)hipMemsetAsync(out, 0, (size_t)out_size * sizeof(float), stream);
        roi_scatter_kernel<true><<<nBlocksScatter, 256, 0, stream>>>(center, feat, out);
    }
}